// HungarianMatcher_55362128445461
// MI455X (gfx1250) — compile-verified
//
#include <hip/hip_runtime.h>
#include <hip/hip_bf16.h>

// Problem dims (fixed by setup_inputs)
#define BS   16
#define NQ   900
#define NC   91
#define NCP  96      // class dim padded to multiple of 4 for WMMA K
#define NT   960
#define BN   (BS * NQ)        // 14400 queries, 900 row tiles of 16
#define TI_TILES (BN / 16)    // 900
#define TJ_TILES (NT / 16)    // 60

#define ALPHA      0.25f
#define COST_CLASS 2.0f
#define COST_BBOX  5.0f
#define COST_GIOU  2.0f
#define EPSF       1e-8f

typedef __attribute__((ext_vector_type(2))) float v2f;
typedef __attribute__((ext_vector_type(8))) float v8f;

// ---------------------------------------------------------------------------
// Kernel 1: per-(query, class) focal cost table, folded cost_class weight.
//   T[i, c] = 2 * ( ALPHA*(1-p)^2*(-log(p+eps)) - (1-ALPHA)*p^2*(-log(1-p+eps)) )
// Padded columns 91..95 are zero (they multiply zero one-hot rows anyway).
// 14400*96*4B = 5.5 MB -> stays resident in the 192 MB L2.
// ---------------------------------------------------------------------------
__global__ void hm_build_class_table(const float* __restrict__ logits,
                                     float* __restrict__ T) {
    int idx = blockIdx.x * blockDim.x + threadIdx.x;
    if (idx >= BN * NCP) return;
    int c = idx % NCP;
    int i = idx / NCP;
    float val = 0.0f;
    if (c < NC) {
        float x = logits[i * NC + c];
        float p = 1.0f / (1.0f + expf(-x));
        float omp = 1.0f - p;
        float pos = ALPHA * omp * omp * (-logf(p + EPSF));
        float neg = (1.0f - ALPHA) * p * p * (-logf(omp + EPSF));
        val = COST_CLASS * (pos - neg);
    }
    T[idx] = val;
}

// ---------------------------------------------------------------------------
// Kernel 2: one wave per 16x16 output tile.
//   Step 1: class cost via 24 chained V_WMMA_F32_16X16X4_F32:
//           acc = T_tile(16x96) x OneHot(96x16), one-hot built in registers.
//   Step 2: same wave adds 5*L1 - 2*GIoU per accumulator element and stores.
// Block = 128 threads = 4 waves, each wave owns one tj tile (60 = 15*4).
// ---------------------------------------------------------------------------
__global__ void __launch_bounds__(128)
hm_cost_kernel(const float* __restrict__ T,
               const float4* __restrict__ pred_boxes,   // [BN] cxcywh
               const int*    __restrict__ tgt_ids,      // [NT]
               const float4* __restrict__ tgt_bbox,     // [NT] cxcywh
               float* __restrict__ out) {               // [BN, NT]
    const int lane = threadIdx.x & 31;
    const int wave = threadIdx.x >> 5;
    const int ti = blockIdx.x;                 // 0..899   (query tile)
    const int tj = blockIdx.y * 4 + wave;      // 0..59    (target tile)

    const int n     = lane & 15;               // tile column handled by lane
    const int khalf = (lane >> 4) << 1;        // 0 for lanes 0-15, 2 for 16-31
    const int mbase = (lane >> 4) << 3;        // C-layout M offset: 0 or 8

    // ---- class gather as one-hot matmul on the matrix unit ----------------
    const float* Trow = T + (size_t)(ti * 16 + (lane & 15)) * NCP;
    const int myid = tgt_ids[tj * 16 + n];     // column n's target class id

    v8f acc = {};
#pragma unroll
    for (int k4 = 0; k4 < NCP; k4 += 4) {
        v2f a, b;
        // A 16x4 f32 fragment: lanes 0-15 -> K=k4,k4+1 ; lanes 16-31 -> K=k4+2,k4+3
        a.x = Trow[k4 + khalf];
        a.y = Trow[k4 + khalf + 1];
        // B 4x16 one-hot fragment, mirrored K split across lane halves
        b.x = (myid == (k4 + khalf))     ? 1.0f : 0.0f;
        b.y = (myid == (k4 + khalf + 1)) ? 1.0f : 0.0f;
        // args: (neg_a, A, neg_b, B, c_mod, C, reuse_a, reuse_b)
        acc = __builtin_amdgcn_wmma_f32_16x16x4_f32(
            false, a, false, b, (short)0, acc, false, false);
    }
    // acc[v] == 2 * cost_class for element (M = mbase+v, N = n)

    // ---- per-lane target box (fixed column): cxcywh -> xyxy + area --------
    const int   tgt = tj * 16 + n;
    const float4 tb = tgt_bbox[tgt];
    const float tx0 = tb.x - 0.5f * tb.z, ty0 = tb.y - 0.5f * tb.w;
    const float tx1 = tb.x + 0.5f * tb.z, ty1 = tb.y + 0.5f * tb.w;
    const float tarea = (tx1 - tx0) * (ty1 - ty0);

    // ---- bbox L1 + GIoU per accumulator row, fused store ------------------
#pragma unroll
    for (int v = 0; v < 8; ++v) {
        const int q = ti * 16 + mbase + v;     // global query index
        const float4 qb = pred_boxes[q];       // broadcast across 16 lanes -> L0 hit

        const float l1 = fabsf(qb.x - tb.x) + fabsf(qb.y - tb.y) +
                         fabsf(qb.z - tb.z) + fabsf(qb.w - tb.w);

        const float qx0 = qb.x - 0.5f * qb.z, qy0 = qb.y - 0.5f * qb.w;
        const float qx1 = qb.x + 0.5f * qb.z, qy1 = qb.y + 0.5f * qb.w;
        const float qarea = (qx1 - qx0) * (qy1 - qy0);

        const float iw = fmaxf(fminf(qx1, tx1) - fmaxf(qx0, tx0), 0.0f);
        const float ih = fmaxf(fminf(qy1, ty1) - fmaxf(qy0, ty0), 0.0f);
        const float inter = iw * ih;
        const float uni   = qarea + tarea - inter;
        const float iou   = inter / uni;

        const float cw = fmaxf(fmaxf(qx1, tx1) - fminf(qx0, tx0), 0.0f);
        const float ch = fmaxf(fmaxf(qy1, ty1) - fminf(qy0, ty0), 0.0f);
        const float areac = cw * ch;
        const float giou  = iou - (areac - uni) / areac;

        const float total = fmaf(COST_BBOX, l1, acc[v]) - COST_GIOU * giou;
        // Non-temporal: don't let the 55 MB output stream evict the L2 table.
        __builtin_nontemporal_store(total, out + (size_t)q * NT + tgt);
    }
}

extern "C" void kernel_launch(void* const* d_in, const int* in_sizes, int n_in,
                              void* d_out, int out_size, void* d_ws, size_t ws_size,
                              hipStream_t stream) {
    const float*  pred_logits = (const float*)d_in[0];  // [16,900,91]
    const float4* pred_boxes  = (const float4*)d_in[1]; // [16,900,4]
    const int*    tgt_ids     = (const int*)d_in[2];    // [960]
    const float4* tgt_bbox    = (const float4*)d_in[3]; // [960,4]
    float* out = (float*)d_out;                         // [16,900,960]
    float* T   = (float*)d_ws;                          // [14400,96] = 5.5 MB

    {
        int total = BN * NCP;
        hm_build_class_table<<<(total + 255) / 256, 256, 0, stream>>>(pred_logits, T);
    }
    {
        dim3 grid(TI_TILES, TJ_TILES / 4);   // (900, 15)
        hm_cost_kernel<<<grid, 128, 0, stream>>>(T, pred_boxes, tgt_ids, tgt_bbox, out);
    }
}